// Reduce_85933705658999
// MI455X (gfx1250) — compile-verified
//
#include <hip/hip_runtime.h>
#include <math.h>
#include <stdint.h>

// ---------------------------------------------------------------------------
// Top-9-along-N mean reduction: x[64][8192][64] f32 -> out[64][64] f32.
// Memory-bound (128 MiB @ 23.3 TB/s ~ 5.5us). No matmul structure -> no WMMA.
// CDNA5 path used: async global->LDS DMA pipeline (ASYNCcnt, depth 8 = 32KB
// in flight per block per Little's-law sizing), wave32-coalesced b128 loads.
// ---------------------------------------------------------------------------

#define TOPK 9
static constexpr int kB     = 64;
static constexpr int kN     = 8192;
static constexpr int kC     = 64;
static constexpr int kSEG   = 8;            // segments per batch (phase 1 grid = 512 blocks)
static constexpr int kRG    = 16;           // row-groups per block
static constexpr int kCQ    = 16;           // channel quads (4 channels each)
static constexpr int kBlock = kRG * kCQ;    // 256 threads = 8 waves (wave32)
static constexpr int kDepth = 8;            // async pipeline depth (power of two)

static_assert(kBlock == 256, "block layout assumption");
static_assert((kDepth & (kDepth - 1)) == 0, "depth must be pow2");
// async stage must fit inside the (aliased) merge-lists LDS region
static_assert(kDepth * kBlock * 4 <= kC * kRG * 9, "stage must fit lists region");

#if defined(__gfx1250__) &&                                              \
    __has_builtin(__builtin_amdgcn_global_load_async_to_lds_b128) &&     \
    __has_builtin(__builtin_amdgcn_s_wait_asynccnt)
#define USE_ASYNC_LDS 1
#else
#define USE_ASYNC_LDS 0
#endif

#if USE_ASYNC_LDS
// Toolchain signature (from diagnostics): (v4i as1*, v4i as3*, Imm int, Imm int)
typedef int v4i __attribute__((vector_size(4 * sizeof(int))));
typedef __attribute__((address_space(1))) v4i gas_v4i;
typedef __attribute__((address_space(3))) v4i las_v4i;
// Flat->LDS addrspacecast is a 32-bit truncation on AMDGPU; flat->global is
// numerically identity. Go through integers so the casts are always legal C++.
__device__ __forceinline__ void async_copy_b128(const void* g, void* l) {
  __builtin_amdgcn_global_load_async_to_lds_b128(
      (gas_v4i*)(unsigned long long)(uintptr_t)g,
      (las_v4i*)(unsigned int)(uintptr_t)l,
      /*offset=*/0, /*cpol=*/0);
}
#endif

// v[0..8] sorted ascending; precondition: x > v[0].
__device__ __forceinline__ void insert9(float (&v)[9], float x) {
#pragma unroll
  for (int i = 0; i < 8; ++i) {
    if (x > v[i + 1]) {
      v[i] = v[i + 1];
    } else {
      v[i] = x;
      return;
    }
  }
  v[8] = x;
}

__device__ __forceinline__ void tryinsert9(float (&v)[9], float x) {
  if (x > v[0]) insert9(v, x);
}

__device__ __forceinline__ void init9(float (&v)[9]) {
#pragma unroll
  for (int k = 0; k < 9; ++k) v[k] = -INFINITY;
}

// Merge a sorted-ascending 9-list from LDS into v (iterate descending, early break).
__device__ __forceinline__ void merge9(float (&v)[9], const float* src) {
#pragma unroll
  for (int k = 8; k >= 0; --k) {
    float val = src[k];
    if (val <= v[0]) break;
    insert9(v, val);
  }
}

// ---------------------------------------------------------------------------
// Phase 1: each block handles (batch b, segment seg) = rows [seg*N/segs, ...).
// Thread t: channel-quad cq = t%16 (channels 4cq..4cq+3), row-group rg = t/16.
// WRITE_OUT=true  -> segs==1, final mean written directly to out.
// WRITE_OUT=false -> per-(b,seg,c) sorted top-9 written to part.
// ---------------------------------------------------------------------------
template <bool WRITE_OUT>
__global__ __launch_bounds__(kBlock) void topk_partial(
    const float* __restrict__ x, float* __restrict__ part,
    float* __restrict__ out, int segs) {
  // lists region: [C][RG][9] = 9216 floats; lvl1 region: [C][4][9] = 2304 floats.
  // async stage (kDepth*kBlock float4 = 8192 floats) aliases the lists region.
  __shared__ __align__(16) float smem[kC * kRG * 9 + kC * 4 * 9];

  const int tid = threadIdx.x;
  const int cq  = tid & 15;
  const int rg  = tid >> 4;

  const int b   = blockIdx.x / segs;
  const int seg = blockIdx.x - b * segs;
  const int rowsPerBlock = kN / segs;
  const int steps        = rowsPerBlock / kRG;   // 64 (segs=8) or 512 (segs=1)

  const float* gbase =
      x + ((size_t)b * kN + (size_t)seg * rowsPerBlock + rg) * kC + cq * 4;
  const size_t stride = (size_t)kRG * kC;        // floats per step (4 KiB/block-step)

  float v0[9], v1[9], v2[9], v3[9];
  init9(v0); init9(v1); init9(v2); init9(v3);

#if USE_ASYNC_LDS
  float4* stage = reinterpret_cast<float4*>(smem);
  auto consume = [&](int i) {
    float4 d = stage[(i & (kDepth - 1)) * kBlock + tid];
    tryinsert9(v0, d.x); tryinsert9(v1, d.y);
    tryinsert9(v2, d.z); tryinsert9(v3, d.w);
  };
#pragma unroll
  for (int p = 0; p < kDepth - 1; ++p)           // steps >= 64 > kDepth-1 always
    async_copy_b128(gbase + (size_t)p * stride, &stage[p * kBlock + tid]);

  int i = 0;
  for (; i < steps - (kDepth - 1); ++i) {
    const int pre = i + kDepth - 1;
    async_copy_b128(gbase + (size_t)pre * stride,
                    &stage[(pre & (kDepth - 1)) * kBlock + tid]);
    __builtin_amdgcn_s_wait_asynccnt(kDepth - 1);
    asm volatile("" ::: "memory");
    consume(i);
  }
  // Epilogue: compile-time-constant wait counts for the last kDepth-1 steps.
  __builtin_amdgcn_s_wait_asynccnt(6); asm volatile("" ::: "memory"); consume(i++);
  __builtin_amdgcn_s_wait_asynccnt(5); asm volatile("" ::: "memory"); consume(i++);
  __builtin_amdgcn_s_wait_asynccnt(4); asm volatile("" ::: "memory"); consume(i++);
  __builtin_amdgcn_s_wait_asynccnt(3); asm volatile("" ::: "memory"); consume(i++);
  __builtin_amdgcn_s_wait_asynccnt(2); asm volatile("" ::: "memory"); consume(i++);
  __builtin_amdgcn_s_wait_asynccnt(1); asm volatile("" ::: "memory"); consume(i++);
  __builtin_amdgcn_s_wait_asynccnt(0); asm volatile("" ::: "memory"); consume(i++);
#else
  for (int i = 0; i < steps; ++i) {
    const float* p = gbase + (size_t)i * stride;
    __builtin_prefetch(p + (size_t)kDepth * stride, 0, 0);  // global_prefetch_b8
    float4 d = *reinterpret_cast<const float4*>(p);
    tryinsert9(v0, d.x); tryinsert9(v1, d.y);
    tryinsert9(v2, d.z); tryinsert9(v3, d.w);
  }
#endif

  // lists region aliases the async stage -> barrier before overwriting.
  __syncthreads();

  const int c0 = cq * 4;
#pragma unroll
  for (int k = 0; k < 9; ++k) {
    smem[((c0 + 0) * kRG + rg) * 9 + k] = v0[k];
    smem[((c0 + 1) * kRG + rg) * 9 + k] = v1[k];
    smem[((c0 + 2) * kRG + rg) * 9 + k] = v2[k];
    smem[((c0 + 3) * kRG + rg) * 9 + k] = v3[k];
  }
  __syncthreads();

  // Level 1: 4 threads per channel, each merges 4 row-group lists.
  float* lvl1 = smem + kC * kRG * 9;
  {
    const int c = tid & 63;
    const int q = tid >> 6;
    float m[9];
    init9(m);
#pragma unroll
    for (int j = 0; j < 4; ++j)
      merge9(m, &smem[(c * kRG + (q * 4 + j)) * 9]);
#pragma unroll
    for (int k = 0; k < 9; ++k) lvl1[(c * 4 + q) * 9 + k] = m[k];
  }
  __syncthreads();

  // Level 2: one thread per channel merges the 4 level-1 lists.
  if (tid < kC) {
    const int c = tid;
    float f[9];
    init9(f);
#pragma unroll
    for (int q = 0; q < 4; ++q) merge9(f, &lvl1[(c * 4 + q) * 9]);

    if (WRITE_OUT) {
      float s = 0.f;
#pragma unroll
      for (int k = 0; k < 9; ++k) s += f[k];
      out[b * kC + c] = s * (1.0f / 9.0f);
    } else {
      float* dst = part + ((size_t)(b * segs + seg) * kC + c) * 9;
#pragma unroll
      for (int k = 0; k < 9; ++k) dst[k] = f[k];
    }
  }
}

// ---------------------------------------------------------------------------
// Phase 2: merge `segs` sorted top-9 partials per (b,c), write mean.
// ---------------------------------------------------------------------------
__global__ __launch_bounds__(kC) void topk_final(const float* __restrict__ part,
                                                 float* __restrict__ out,
                                                 int segs) {
  const int b = blockIdx.x;
  const int c = threadIdx.x;
  float v[9];
  init9(v);
  for (int s = 0; s < segs; ++s)
    merge9(v, part + ((size_t)(b * segs + s) * kC + c) * 9);
  float sum = 0.f;
#pragma unroll
  for (int k = 0; k < 9; ++k) sum += v[k];
  out[b * kC + c] = sum * (1.0f / 9.0f);
}

extern "C" void kernel_launch(void* const* d_in, const int* in_sizes, int n_in,
                              void* d_out, int out_size, void* d_ws,
                              size_t ws_size, hipStream_t stream) {
  (void)in_sizes; (void)n_in; (void)out_size;
  const float* x = (const float*)d_in[0];
  float* out = (float*)d_out;

  const size_t need = (size_t)kB * kSEG * kC * TOPK * sizeof(float);  // ~1.1 MB
  if (d_ws != nullptr && ws_size >= need) {
    float* part = (float*)d_ws;
    topk_partial<false><<<dim3(kB * kSEG), dim3(kBlock), 0, stream>>>(
        x, part, nullptr, kSEG);
    topk_final<<<dim3(kB), dim3(kC), 0, stream>>>(part, out, kSEG);
  } else {
    // Fused fallback (no workspace): 64 blocks, each does a full batch.
    topk_partial<true><<<dim3(kB), dim3(kBlock), 0, stream>>>(
        x, nullptr, out, 1);
  }
}